// SpatialTransformer2_10625749090618
// MI455X (gfx1250) — compile-verified
//
#include <hip/hip_runtime.h>

// SpatialTransformer (3D trilinear warp) for MI455X / gfx1250.
//
// src:  (1, 2, 160, 192, 224) fp32  -> d_in[0], laid out as (2, N)
// flow: (1, 3, 160, 192, 224) fp32  -> d_in[1], laid out as (3, N)
// out:  (1, 2, 160, 192, 224) fp32  -> d_out,   laid out as (2, N)
//
// Memory-bound (~193 MB / 23.3 TB/s ~= 8.3 us floor). gfx1250 strategy:
//  - NT temporal hints on the single-pass flow/out streams so the 55 MB src
//    volume stays resident in the 192 MB L2 for the gathers.
//  - global_prefetch_b8 ahead on the flow streams.
//  - Per-axis hoisted clamp/validity (6 instead of 24 each) with validity
//    folded into axis weights (exactly reproduces grid_sample zeros padding,
//    align_corners=True).
//  - One base flat index + {0,HW}/{0,W}/{0,1} delta tree -> 7 adds for all
//    8 corner addresses; both channels gather through separate uniform SGPR
//    bases with shared VGPR offsets.

typedef float v4f __attribute__((ext_vector_type(4)));

constexpr int kD = 160;
constexpr int kH = 192;
constexpr int kW = 224;
constexpr int kHW = kH * kW;               // 43,008
constexpr int kN = kD * kH * kW;           // 6,881,280 voxels per channel
constexpr int kPrefetchDist = 4096;        // floats (~16 KB) ahead per stream

__global__ __launch_bounds__(256)
void st3d_trilinear_warp(const float* __restrict__ src,
                         const float* __restrict__ flow,
                         float* __restrict__ out) {
  const int t = blockIdx.x * blockDim.x + threadIdx.x;
  const int base = t * 4;                  // 4 consecutive w positions
  if (base >= kN) return;

  const float* __restrict__ src1 = src + kN;   // channel 1, uniform SGPR base

  // Prefetch the three flow streams ahead (speculative; OOB is dropped).
  __builtin_prefetch(flow + base + kPrefetchDist);
  __builtin_prefetch(flow + kN + base + kPrefetchDist);
  __builtin_prefetch(flow + 2 * kN + base + kPrefetchDist);

  // Decompose linear index -> (dz, hy, wcol). W % 4 == 0, so the 4-vector
  // never crosses a row boundary; dz/hy are uniform across the vector.
  const int wcol = base % kW;
  const int rest = base / kW;
  const int hy = rest % kH;
  const int dz = rest / kH;

  // Non-temporal streaming loads of the flow (read exactly once).
  const v4f fD = __builtin_nontemporal_load((const v4f*)(flow + base));
  const v4f fH = __builtin_nontemporal_load((const v4f*)(flow + kN + base));
  const v4f fW = __builtin_nontemporal_load((const v4f*)(flow + 2 * kN + base));

  const float dzf = (float)dz;
  const float hyf = (float)hy;

  v4f r0, r1;

#pragma unroll
  for (int j = 0; j < 4; ++j) {
    // Voxel-space sample coordinates (align_corners=True: grid + flow).
    const float dco = dzf + fD[j];
    const float hco = hyf + fH[j];
    const float wco = (float)(wcol + j) + fW[j];

    const float dfl = floorf(dco);
    const float hfl = floorf(hco);
    const float wfl = floorf(wco);

    const float fd = dco - dfl;
    const float fh = hco - hfl;
    const float fw = wco - wfl;

    const int d0 = (int)dfl, d1 = d0 + 1;
    const int h0 = (int)hfl, h1 = h0 + 1;
    const int w0 = (int)wfl, w1 = w0 + 1;

    // Per-axis clamped indices (loads are always in-bounds).
    const int dc0 = min(max(d0, 0), kD - 1);
    const int dc1 = min(max(d1, 0), kD - 1);
    const int hc0 = min(max(h0, 0), kH - 1);
    const int hc1 = min(max(h1, 0), kH - 1);
    const int wc0 = min(max(w0, 0), kW - 1);
    const int wc1 = min(max(w1, 0), kW - 1);

    // Per-axis weights with validity folded in (zeros-padding semantics:
    // any invalid axis factor zeroes the whole corner weight).
    const float wd0 = ((unsigned)d0 < (unsigned)kD) ? (1.0f - fd) : 0.0f;
    const float wd1 = ((unsigned)d1 < (unsigned)kD) ? fd : 0.0f;
    const float wh0 = ((unsigned)h0 < (unsigned)kH) ? (1.0f - fh) : 0.0f;
    const float wh1 = ((unsigned)h1 < (unsigned)kH) ? fh : 0.0f;
    const float ww0 = ((unsigned)w0 < (unsigned)kW) ? (1.0f - fw) : 0.0f;
    const float ww1 = ((unsigned)w1 < (unsigned)kW) ? fw : 0.0f;

    // Corner weights (reference order: dd, hh, ww).
    const float c000 = wd0 * (wh0 * ww0);
    const float c001 = wd0 * (wh0 * ww1);
    const float c010 = wd0 * (wh1 * ww0);
    const float c011 = wd0 * (wh1 * ww1);
    const float c100 = wd1 * (wh0 * ww0);
    const float c101 = wd1 * (wh0 * ww1);
    const float c110 = wd1 * (wh1 * ww0);
    const float c111 = wd1 * (wh1 * ww1);

    // One base address + delta tree for all 8 corners.
    const int fbase = (dc0 * kH + hc0) * kW + wc0;
    const int dd = (dc1 - dc0) * kHW;      // 0 or HW
    const int dh = (hc1 - hc0) * kW;       // 0 or W
    const int dw = wc1 - wc0;              // 0 or 1

    const int f000 = fbase;
    const int f001 = fbase + dw;
    const int f010 = fbase + dh;
    const int f011 = f010 + dw;
    const int f100 = f000 + dd;
    const int f101 = f001 + dd;
    const int f110 = f010 + dd;
    const int f111 = f011 + dd;

    // Channel 0 and channel 1 share VGPR offsets against separate SGPR bases.
    float a0 = src[f000] * c000;
    float a1 = src1[f000] * c000;
    a0 = fmaf(src[f001], c001, a0);  a1 = fmaf(src1[f001], c001, a1);
    a0 = fmaf(src[f010], c010, a0);  a1 = fmaf(src1[f010], c010, a1);
    a0 = fmaf(src[f011], c011, a0);  a1 = fmaf(src1[f011], c011, a1);
    a0 = fmaf(src[f100], c100, a0);  a1 = fmaf(src1[f100], c100, a1);
    a0 = fmaf(src[f101], c101, a0);  a1 = fmaf(src1[f101], c101, a1);
    a0 = fmaf(src[f110], c110, a0);  a1 = fmaf(src1[f110], c110, a1);
    a0 = fmaf(src[f111], c111, a0);  a1 = fmaf(src1[f111], c111, a1);

    r0[j] = a0;
    r1[j] = a1;
  }

  // Non-temporal streaming stores (output is write-once, never re-read).
  __builtin_nontemporal_store(r0, (v4f*)(out + base));
  __builtin_nontemporal_store(r1, (v4f*)(out + kN + base));
}

extern "C" void kernel_launch(void* const* d_in, const int* in_sizes, int n_in,
                              void* d_out, int out_size, void* d_ws, size_t ws_size,
                              hipStream_t stream) {
  const float* src = (const float*)d_in[0];   // (2, N) fp32
  const float* flow = (const float*)d_in[1];  // (3, N) fp32
  float* out = (float*)d_out;                 // (2, N) fp32

  constexpr int kVec = kN / 4;                // threads (4 voxels each)
  const dim3 block(256);
  const dim3 grid((kVec + 255) / 256);
  st3d_trilinear_warp<<<grid, block, 0, stream>>>(src, flow, out);
}